// CBAM_54898271977497
// MI455X (gfx1250) — compile-verified
//
#include <hip/hip_runtime.h>

typedef __attribute__((ext_vector_type(2))) float v2f;
typedef __attribute__((ext_vector_type(8))) float v8f;

#define Bn  32
#define Cn  256
#define Rr  16
#define Hn  64
#define HWn 4096

__device__ __forceinline__ float sigmoidf(float v) {
    return 1.0f / (1.0f + expf(-v));
}

// ---------------------------------------------------------------------------
// Kernel A: per-(b,c) spatial mean & max over 64x64. One block per (b,c).
// Writes stacked A[64,256]: rows 0..31 = avg (per batch), rows 32..63 = max.
// ---------------------------------------------------------------------------
__global__ void __launch_bounds__(256) cbam_stats(const float* __restrict__ x,
                                                  float* __restrict__ A) {
    const int bc  = blockIdx.x;            // b*256 + c
    const int tid = threadIdx.x;
    const float4* p4 = (const float4*)(x + (size_t)bc * HWn);

    float s = 0.0f;
    float m = -__builtin_inff();
#pragma unroll
    for (int j = 0; j < 4; ++j) {          // 4096 floats = 1024 float4
        float4 v = p4[j * 256 + tid];
        s += v.x + v.y + v.z + v.w;
        m = fmaxf(m, fmaxf(fmaxf(v.x, v.y), fmaxf(v.z, v.w)));
    }

    __shared__ float ss[256];
    __shared__ float sm[256];
    ss[tid] = s; sm[tid] = m;
    __syncthreads();
    for (int off = 128; off > 0; off >>= 1) {
        if (tid < off) {
            ss[tid] += ss[tid + off];
            sm[tid]  = fmaxf(sm[tid], sm[tid + off]);
        }
        __syncthreads();
    }
    if (tid == 0) {
        const int b = bc >> 8, c = bc & 255;
        A[b * Cn + c]        = ss[0] * (1.0f / (float)HWn);  // avg rows 0..31
        A[(b + Bn) * Cn + c] = sm[0];                        // max rows 32..63
    }
}

// ---------------------------------------------------------------------------
// Kernel B: shared MLP via V_WMMA_F32_16X16X4_F32, single wave (EXEC all 1s).
//   h[64,16]  = relu(A[64,256] @ w1^T)     w1: [16,256]  B[k][n] = w1[n*256+k]
//   o[64,256] = h @ w2^T                   w2: [256,16]  B[k][n] = w2[n*16+k]
//   ch_att[b,c] = sigmoid(o[b,c] + o[b+32,c])  (avg-path + max-path rows)
// Fragment layouts per CDNA5 ISA 7.12.2:
//   A 16x4 : lane m=lane%16, VGPR0/1 hold K = k0+2*(lane/16) + {0,1}
//   B 4x16 : VGPR v <-> K=k0+2*(lane/16)+v; D 16x16 : VGPR v <-> M=v+8*(lane/16)
// ---------------------------------------------------------------------------
__global__ void __launch_bounds__(32) cbam_mlp_wmma(const float* __restrict__ A,
                                                    const float* __restrict__ w1,
                                                    const float* __restrict__ w2,
                                                    float* __restrict__ ch_att) {
    const int lane = threadIdx.x;
    const int ll   = lane & 15;   // M (for A) / N (for B,D) index
    const int lh   = lane >> 4;   // half-wave selector

    __shared__ float hbuf[64 * 16];   // relu'd hidden activations

    // ---- GEMM1: 4 M-tiles x 64 K-steps ----
    for (int mt = 0; mt < 4; ++mt) {
        v8f acc = {0.f, 0.f, 0.f, 0.f, 0.f, 0.f, 0.f, 0.f};
        const float* arow = A + (mt * 16 + ll) * Cn;
        for (int k0 = 0; k0 < Cn; k0 += 4) {
            const int k = k0 + 2 * lh;
            v2f a, b;
            a.x = arow[k];
            a.y = arow[k + 1];
            b.x = w1[ll * Cn + k];        // B[k][n]=w1[n][k], n=ll
            b.y = w1[ll * Cn + k + 1];
            acc = __builtin_amdgcn_wmma_f32_16x16x4_f32(
                false, a, false, b, (short)0, acc, false, false);
        }
#pragma unroll
        for (int v = 0; v < 8; ++v)       // D: M = mt*16 + v + 8*lh, N = ll
            hbuf[(mt * 16 + v + 8 * lh) * Rr + ll] = fmaxf(acc[v], 0.0f);
    }
    __syncthreads();

    // ---- GEMM2: avg-path (rows mt*16..) and max-path (rows (mt+2)*16..)
    //      share B fragments and lane layout -> fuse add+sigmoid directly ----
    for (int mt = 0; mt < 2; ++mt) {
        for (int nt = 0; nt < 16; ++nt) {
            v8f acc0 = {0.f, 0.f, 0.f, 0.f, 0.f, 0.f, 0.f, 0.f};
            v8f acc1 = {0.f, 0.f, 0.f, 0.f, 0.f, 0.f, 0.f, 0.f};
            for (int k0 = 0; k0 < Rr; k0 += 4) {
                const int k = k0 + 2 * lh;
                v2f a0, a1, b;
                a0.x = hbuf[(mt * 16 + ll) * Rr + k];
                a0.y = hbuf[(mt * 16 + ll) * Rr + k + 1];
                a1.x = hbuf[((mt + 2) * 16 + ll) * Rr + k];
                a1.y = hbuf[((mt + 2) * 16 + ll) * Rr + k + 1];
                b.x  = w2[(nt * 16 + ll) * Rr + k];   // B[k][n]=w2[n][k]
                b.y  = w2[(nt * 16 + ll) * Rr + k + 1];
                acc0 = __builtin_amdgcn_wmma_f32_16x16x4_f32(
                    false, a0, false, b, (short)0, acc0, false, false);
                acc1 = __builtin_amdgcn_wmma_f32_16x16x4_f32(
                    false, a1, false, b, (short)0, acc1, false, false);
            }
#pragma unroll
            for (int v = 0; v < 8; ++v) {
                const int row = mt * 16 + v + 8 * lh;   // batch 0..31
                const int col = nt * 16 + ll;           // channel
                ch_att[row * Cn + col] = sigmoidf(acc0[v] + acc1[v]);
            }
        }
    }
}

// ---------------------------------------------------------------------------
// Kernel C: per-pixel channel mean/max of (x * ch_att). One block = one batch
// slice of 256 contiguous pixels; thread owns one pixel, loops channels
// (stride HWn -> each iteration is a fully coalesced 1KB load per block).
// ---------------------------------------------------------------------------
__global__ void __launch_bounds__(256) cbam_spstats(const float* __restrict__ x,
                                                    const float* __restrict__ ch_att,
                                                    float* __restrict__ avg_sp,
                                                    float* __restrict__ max_sp) {
    const int b  = blockIdx.x >> 4;                 // 16 blocks per batch
    const int hw = ((blockIdx.x & 15) << 8) + threadIdx.x;

    __shared__ float sa[Cn];
    sa[threadIdx.x] = ch_att[b * Cn + threadIdx.x];
    __syncthreads();

    const float* p = x + (size_t)b * Cn * HWn + hw;
    float s = 0.0f;
    float m = -__builtin_inff();
    for (int c = 0; c < Cn; ++c) {
        float v = p[(size_t)c * HWn] * sa[c];
        s += v;
        m = fmaxf(m, v);
    }
    avg_sp[b * HWn + hw] = s * (1.0f / (float)Cn);
    max_sp[b * HWn + hw] = m;
}

// ---------------------------------------------------------------------------
// Kernel D: 7x7 'SAME' conv over [avg_sp ; max_sp] + sigmoid -> s_map[B,HW].
// w_sp OIHW [1,2,7,7]: [0..48] = avg-channel taps, [49..97] = max-channel.
// ---------------------------------------------------------------------------
__global__ void __launch_bounds__(256) cbam_spconv(const float* __restrict__ avg_sp,
                                                   const float* __restrict__ max_sp,
                                                   const float* __restrict__ w_sp,
                                                   float* __restrict__ s_out) {
    const int idx = blockIdx.x * 256 + threadIdx.x;   // b*HWn + hw
    const int b   = idx >> 12;
    const int hw  = idx & (HWn - 1);
    const int h   = hw >> 6, w = hw & 63;

    __shared__ float wk[98];
    if (threadIdx.x < 98) wk[threadIdx.x] = w_sp[threadIdx.x];
    __syncthreads();

    const float* pa = avg_sp + b * HWn;
    const float* pm = max_sp + b * HWn;
    float acc = 0.0f;
#pragma unroll
    for (int i = 0; i < 7; ++i) {
        const int hh = h + i - 3;
        if (hh < 0 || hh >= Hn) continue;
#pragma unroll
        for (int j = 0; j < 7; ++j) {
            const int ww = w + j - 3;
            if (ww < 0 || ww >= Hn) continue;
            acc += wk[i * 7 + j]      * pa[hh * Hn + ww]
                 + wk[49 + i * 7 + j] * pm[hh * Hn + ww];
        }
    }
    s_out[idx] = sigmoidf(acc);
}

// ---------------------------------------------------------------------------
// Kernel E: out = x * ch_att[b,c] * s_map[b,hw], float4 vectorized (b128).
// ---------------------------------------------------------------------------
__global__ void __launch_bounds__(256) cbam_apply(const float4* __restrict__ x4,
                                                  const float* __restrict__ ch_att,
                                                  const float* __restrict__ s_map,
                                                  float4* __restrict__ out4) {
    const size_t i4  = (size_t)blockIdx.x * 256 + threadIdx.x;
    const size_t bc  = i4 >> 10;              // 1024 float4 per (b,c) plane
    const int    b   = (int)(bc >> 8);
    const int    hw4 = (int)(i4 & 1023);

    const float  a  = ch_att[bc];
    const float4 xv = x4[i4];
    const float* sp = s_map + b * HWn + hw4 * 4;

    float4 o;
    o.x = xv.x * a * sp[0];
    o.y = xv.y * a * sp[1];
    o.z = xv.z * a * sp[2];
    o.w = xv.w * a * sp[3];
    out4[i4] = o;
}

// ---------------------------------------------------------------------------
extern "C" void kernel_launch(void* const* d_in, const int* in_sizes, int n_in,
                              void* d_out, int out_size, void* d_ws, size_t ws_size,
                              hipStream_t stream) {
    (void)in_sizes; (void)n_in; (void)out_size; (void)ws_size;

    const float* x    = (const float*)d_in[0];   // [32,256,64,64]
    const float* w1   = (const float*)d_in[1];   // [16,256]
    const float* w2   = (const float*)d_in[2];   // [256,16]
    const float* w_sp = (const float*)d_in[3];   // [1,2,7,7]
    float*       out  = (float*)d_out;

    char* ws = (char*)d_ws;
    float* A      = (float*)(ws);                 // [64,256]  stacked avg|max
    float* ch_att = (float*)(ws + 65536);         // [32,256]
    float* avg_sp = (float*)(ws + 98304);         // [32,4096]
    float* max_sp = (float*)(ws + 622592);        // [32,4096]
    float* s_map  = (float*)(ws + 1146880);       // [32,4096]

    cbam_stats   <<<Bn * Cn,            256, 0, stream>>>(x, A);
    cbam_mlp_wmma<<<1,                   32, 0, stream>>>(A, w1, w2, ch_att);
    cbam_spstats <<<Bn * (HWn / 256),   256, 0, stream>>>(x, ch_att, avg_sp, max_sp);
    cbam_spconv  <<<Bn * HWn / 256,     256, 0, stream>>>(avg_sp, max_sp, w_sp, s_map);
    cbam_apply   <<<(Bn * Cn * HWn / 4) / 256, 256, 0, stream>>>(
        (const float4*)x, ch_att, s_map, (float4*)out);
}